// MambaTradingModel_5789615915605
// MI455X (gfx1250) — compile-verified
//
#include <hip/hip_runtime.h>
#include <hip/hip_bf16.h>
#include <math.h>

#define N_LAYERS 4
#define D_MODEL  256
#define D_INNER  512
#define D_STATE  16
#define D_CONV   4
#define DT_RANK  16
#define NFEAT    32
#define NCLS     3
#define BATCH    8
#define SEQ      2048
#define NTOK     (BATCH*SEQ)           // 16384
#define XD_W     (DT_RANK + 2*D_STATE) // 48 floats = 192 bytes per row
#define CHUNK    8                     // timesteps staged per LDS buffer (= waves/block)

typedef float v2f __attribute__((ext_vector_type(2)));
typedef float v8f __attribute__((ext_vector_type(8)));

#if defined(__has_builtin)
#if __has_builtin(__builtin_amdgcn_wmma_f32_16x16x4_f32)
#define HAVE_WMMA_F32 1
#endif
#endif

__device__ __forceinline__ float sigmoidf_(float x) { return 1.f / (1.f + __expf(-x)); }
__device__ __forceinline__ float softplusf_(float x) {
    return x > 0.f ? x + log1pf(__expf(-x)) : log1pf(__expf(x));
}

__device__ __forceinline__ void wait_async0_() {
#if defined(__has_builtin)
#if __has_builtin(__builtin_amdgcn_s_wait_asynccnt)
    __builtin_amdgcn_s_wait_asynccnt(0);
    return;
#else
    asm volatile("s_wait_asynccnt 0" ::: "memory");
    return;
#endif
#else
    asm volatile("s_wait_asynccnt 0" ::: "memory");
#endif
}

// Async global -> LDS copy of 16 bytes per active lane (GLOBAL_LOAD_ASYNC_TO_LDS_B128,
// GV mode: 64-bit address in VGPR pair).  Tracked by ASYNCcnt.
__device__ __forceinline__ void async_g2l_b128_(uint32_t lds_byte_off, uint64_t gaddr) {
    asm volatile("global_load_async_to_lds_b128 %0, %1, off"
                 :: "v"(lds_byte_off), "v"(gaddr) : "memory");
}

// ---------------------------------------------------------------------------
// C[M,N] = A[M,K] @ W[K,N] (+bias) (+= residual).  One wave per 16x16 C tile.
// A-tile layout (ISA 7.12.2, 32-bit A 16x4): lanes 0-15 rows M=0..15 hold
// K={k0,k0+1}; lanes 16..31 hold K={k0+2,k0+3}.  B 4x16 row-striped across
// lanes with the same half split.  C/D: VGPR p -> row m0+p (lanes 0-15) /
// m0+p+8 (lanes 16-31), col = n0+(lane&15).
// ---------------------------------------------------------------------------
template<bool HAS_BIAS, bool ACCUM>
__global__ __launch_bounds__(128)
void wmma_gemm_f32(const float* __restrict__ A, const float* __restrict__ W,
                   const float* __restrict__ bias, float* __restrict__ C,
                   int M, int N, int K)
{
    const int lane = threadIdx.x;                    // 0..31
    const int n0   = blockIdx.x * 16;
    const int m0   = (blockIdx.y * 4 + threadIdx.y) * 16;
    const int half = lane >> 4;
    const int l15  = lane & 15;
    const int col  = n0 + l15;

    v8f acc = {};
#if HAVE_WMMA_F32
    const float* arow = A + (size_t)(m0 + l15) * K;
    const float* wcol = W + col;
    for (int k0 = 0; k0 < K; k0 += 4) {
        const int ka = k0 + 2 * half;
        v2f a, b;
        a.x = arow[ka];
        a.y = arow[ka + 1];
        b.x = wcol[(size_t)ka * N];
        b.y = wcol[(size_t)(ka + 1) * N];
        acc = __builtin_amdgcn_wmma_f32_16x16x4_f32(
            /*neg_a=*/false, a, /*neg_b=*/false, b,
            /*c_mod=*/(short)0, acc, /*reuse_a=*/false, /*reuse_b=*/false);
    }
#else
    #pragma unroll
    for (int p = 0; p < 8; ++p) {
        const float* arow = A + (size_t)(m0 + p + 8 * half) * K;
        float s = 0.f;
        for (int k = 0; k < K; ++k) s += arow[k] * W[(size_t)k * N + col];
        acc[p] = s;
    }
#endif
    const float bv = HAS_BIAS ? bias[col] : 0.f;
    #pragma unroll
    for (int p = 0; p < 8; ++p) {
        const int row = m0 + p + 8 * half;
        const size_t idx = (size_t)row * N + col;
        float v = acc[p] + bv;
        if (ACCUM) v += C[idx];
        C[idx] = v;
    }
}

// ---------------------------------------------------------------------------
// LayerNorm over D_MODEL=256, one wave per token (8 elems/lane), wave32 shuffles.
// ---------------------------------------------------------------------------
__global__ __launch_bounds__(128)
void layernorm_k(const float* __restrict__ x, const float* __restrict__ w,
                 const float* __restrict__ b, float* __restrict__ out)
{
    const int tok  = blockIdx.x * blockDim.y + threadIdx.y;
    const int lane = threadIdx.x;
    const float* xr = x + (size_t)tok * D_MODEL;
    float v[8], s = 0.f;
    #pragma unroll
    for (int i = 0; i < 8; ++i) { v[i] = xr[i * 32 + lane]; s += v[i]; }
    #pragma unroll
    for (int o = 16; o; o >>= 1) s += __shfl_xor(s, o, 32);
    const float mu = s * (1.f / D_MODEL);
    float q = 0.f;
    #pragma unroll
    for (int i = 0; i < 8; ++i) { float d = v[i] - mu; q += d * d; }
    #pragma unroll
    for (int o = 16; o; o >>= 1) q += __shfl_xor(q, o, 32);
    const float rs = rsqrtf(q * (1.f / D_MODEL) + 1e-5f);
    float* orow = out + (size_t)tok * D_MODEL;
    #pragma unroll
    for (int i = 0; i < 8; ++i) {
        const int c = i * 32 + lane;
        orow[c] = (v[i] - mu) * rs * w[c] + b[c];
    }
}

// ---------------------------------------------------------------------------
// Depthwise causal conv (D_CONV=4) over seq + bias + SiLU.  xs = xz[..., :512].
// ---------------------------------------------------------------------------
__global__ __launch_bounds__(256)
void conv_silu_k(const float* __restrict__ xz, const float* __restrict__ cw,
                 const float* __restrict__ cb, float* __restrict__ xc)
{
    const int idx = blockIdx.x * 256 + threadIdx.x;   // over NTOK*D_INNER
    const int d   = idx & (D_INNER - 1);
    const int tok = idx >> 9;
    const int t   = tok & (SEQ - 1);
    float acc = cb[d];
    #pragma unroll
    for (int j = 0; j < D_CONV; ++j) {
        const int tt = t - (D_CONV - 1) + j;
        const float xv = (tt >= 0) ? xz[(size_t)(tok + (tt - t)) * (2 * D_INNER) + d] : 0.f;
        acc += xv * cw[j * D_INNER + d];
    }
    xc[idx] = acc * sigmoidf_(acc);
}

// ---------------------------------------------------------------------------
// SSM recurrence.  One thread per (b, d) channel, 16-state in registers.
// The 48-float xd rows (dt_r | B | C) are streamed global->LDS with
// GLOBAL_LOAD_ASYNC_TO_LDS_B128 in double-buffered chunks of CHUNK=8
// timesteps: wave w stages row w of the *next* chunk (12 lanes x B128) while
// the current chunk is consumed, retiring with s_wait_asynccnt + one barrier
// per chunk instead of two barriers + a synchronous load per timestep.
// dt = softplus(dt_r @ dtW[:,d] + dtB[d]) is fused; the epilogue fuses
// y += x*D and the silu(z) gate, overwriting xc in place.
// ---------------------------------------------------------------------------
__global__ __launch_bounds__(256)
void ssm_scan_k(const float* __restrict__ xd, const float* __restrict__ xz,
                const float* __restrict__ dtW_l, const float* __restrict__ dtB_l,
                const float* __restrict__ Alog_l, const float* __restrict__ D_l,
                float* __restrict__ xc)
{
    __shared__ float rows[2][CHUNK][XD_W];
    const int b    = blockIdx.y;
    const int d    = blockIdx.x * 256 + threadIdx.x;
    const int wave = threadIdx.x >> 5;               // 0..7, stages row `wave`
    const int lane = threadIdx.x & 31;

    float Acoef[D_STATE];
    #pragma unroll
    for (int s = 0; s < D_STATE; ++s) Acoef[s] = -__expf(Alog_l[d * D_STATE + s]);
    float dtw[DT_RANK];
    #pragma unroll
    for (int r = 0; r < DT_RANK; ++r) dtw[r] = dtW_l[r * D_INNER + d];
    const float dtb = dtB_l[d];
    const float Dd  = D_l[d];

    float h[D_STATE];
    #pragma unroll
    for (int s = 0; s < D_STATE; ++s) h[s] = 0.f;

    const size_t tok0 = (size_t)b * SEQ;

    // Preload chunk 0 into buffer 0 (one async B128 instruction per wave).
    if (lane < 12) {
        const uint64_t ga = (uint64_t)(uintptr_t)(xd + (tok0 + wave) * XD_W)
                          + (uint32_t)(lane * 16);
        const uint32_t la = (uint32_t)(uintptr_t)(&rows[0][wave][0])
                          + (uint32_t)(lane * 16);
        async_g2l_b128_(la, ga);
    }
    wait_async0_();
    __syncthreads();

    const int nchunks = SEQ / CHUNK;
    for (int c = 0; c < nchunks; ++c) {
        const int p = c & 1;
        // Stage next chunk into the other buffer (safe: the barrier ending the
        // previous iteration guarantees all waves finished reading it).
        if (c + 1 < nchunks && lane < 12) {
            const uint64_t ga =
                (uint64_t)(uintptr_t)(xd + (tok0 + (size_t)(c + 1) * CHUNK + wave) * XD_W)
                + (uint32_t)(lane * 16);
            const uint32_t la = (uint32_t)(uintptr_t)(&rows[p ^ 1][wave][0])
                              + (uint32_t)(lane * 16);
            async_g2l_b128_(la, ga);
        }

        #pragma unroll
        for (int u = 0; u < CHUNK; ++u) {
            const size_t tok = tok0 + (size_t)c * CHUNK + u;
            const float* row = rows[p][u];

            const float xv = xc[tok * D_INNER + d];
            const float zv = xz[tok * (2 * D_INNER) + D_INNER + d];

            float dt = dtb;
            #pragma unroll
            for (int r = 0; r < DT_RANK; ++r) dt += row[r] * dtw[r];
            dt = softplusf_(dt);

            float y = 0.f;
            #pragma unroll
            for (int s = 0; s < D_STATE; ++s) {
                const float dA = __expf(dt * Acoef[s]);
                h[s] = dA * h[s] + (dt * row[DT_RANK + s]) * xv;
                y += h[s] * row[DT_RANK + D_STATE + s];
            }
            y += xv * Dd;
            xc[tok * D_INNER + d] = y * zv * sigmoidf_(zv);
        }

        wait_async0_();   // this wave's next-chunk row has landed in LDS
        __syncthreads();  // all waves' rows landed; all done with rows[p]
    }
}

// ---------------------------------------------------------------------------
// Final LN of last token + MLP head (256 -> 128 gelu(tanh) -> 3).  1 block / b.
// ---------------------------------------------------------------------------
__global__ __launch_bounds__(128)
void head_k(const float* __restrict__ hbuf, const float* __restrict__ flw,
            const float* __restrict__ flb, const float* __restrict__ h1W,
            const float* __restrict__ h1b, const float* __restrict__ h2W,
            const float* __restrict__ h2b, float* __restrict__ out)
{
    __shared__ float tmp[128];
    __shared__ float ln[D_MODEL];
    __shared__ float hid[D_MODEL / 2];
    const int b = blockIdx.x, tid = threadIdx.x;
    const float* xr = hbuf + (size_t)(b * SEQ + SEQ - 1) * D_MODEL;
    const float a0 = xr[tid], a1 = xr[tid + 128];

    tmp[tid] = a0 + a1; __syncthreads();
    for (int o = 64; o; o >>= 1) { if (tid < o) tmp[tid] += tmp[tid + o]; __syncthreads(); }
    const float mu = tmp[0] * (1.f / D_MODEL); __syncthreads();
    tmp[tid] = (a0 - mu) * (a0 - mu) + (a1 - mu) * (a1 - mu); __syncthreads();
    for (int o = 64; o; o >>= 1) { if (tid < o) tmp[tid] += tmp[tid + o]; __syncthreads(); }
    const float rs = rsqrtf(tmp[0] * (1.f / D_MODEL) + 1e-5f); __syncthreads();

    ln[tid]       = (a0 - mu) * rs * flw[tid] + flb[tid];
    ln[tid + 128] = (a1 - mu) * rs * flw[tid + 128] + flb[tid + 128];
    __syncthreads();

    float acc = h1b[tid];
    for (int k = 0; k < D_MODEL; ++k) acc += ln[k] * h1W[k * (D_MODEL / 2) + tid];
    const float x3 = acc * acc * acc;
    hid[tid] = 0.5f * acc * (1.f + tanhf(0.7978845608028654f * (acc + 0.044715f * x3)));
    __syncthreads();

    if (tid < NCLS) {
        float o = h2b[tid];
        for (int j = 0; j < D_MODEL / 2; ++j) o += hid[j] * h2W[j * NCLS + tid];
        out[b * NCLS + tid] = o;
    }
}

extern "C" void kernel_launch(void* const* d_in, const int* in_sizes, int n_in,
                              void* d_out, int out_size, void* d_ws, size_t ws_size,
                              hipStream_t stream)
{
    const float* x     = (const float*)d_in[0];
    const float* in_w  = (const float*)d_in[1];
    const float* in_b  = (const float*)d_in[2];
    const float* ln_w  = (const float*)d_in[3];
    const float* ln_b  = (const float*)d_in[4];
    const float* inW   = (const float*)d_in[5];
    const float* convW = (const float*)d_in[6];
    const float* convB = (const float*)d_in[7];
    const float* xpW   = (const float*)d_in[8];
    const float* dtW   = (const float*)d_in[9];
    const float* A_log = (const float*)d_in[10];
    const float* Dp    = (const float*)d_in[11];
    const float* outW  = (const float*)d_in[12];
    const float* fln_w = (const float*)d_in[13];
    const float* fln_b = (const float*)d_in[14];
    const float* h1W   = (const float*)d_in[15];
    const float* h1b   = (const float*)d_in[16];
    const float* h2W   = (const float*)d_in[17];
    const float* h2b   = (const float*)d_in[18];
    const float* dtB   = (const float*)d_in[19];
    float* out = (float*)d_out;

    float* ws   = (float*)d_ws;
    float* hbuf = ws;                                   // NTOK*256
    float* xln  = hbuf + (size_t)NTOK * D_MODEL;        // NTOK*256
    float* xz   = xln  + (size_t)NTOK * D_MODEL;        // NTOK*1024
    float* xc   = xz   + (size_t)NTOK * 2 * D_INNER;    // NTOK*512 (conv out, then gated y)
    float* xd   = xc   + (size_t)NTOK * D_INNER;        // NTOK*48

    const dim3 wblk(32, 4);

    // h = x @ in_w + in_b
    wmma_gemm_f32<true, false><<<dim3(D_MODEL / 16, NTOK / 64), wblk, 0, stream>>>(
        x, in_w, in_b, hbuf, NTOK, D_MODEL, NFEAT);

    for (int i = 0; i < N_LAYERS; ++i) {
        layernorm_k<<<NTOK / 4, dim3(32, 4), 0, stream>>>(
            hbuf, ln_w + i * D_MODEL, ln_b + i * D_MODEL, xln);

        wmma_gemm_f32<false, false><<<dim3(2 * D_INNER / 16, NTOK / 64), wblk, 0, stream>>>(
            xln, inW + (size_t)i * D_MODEL * 2 * D_INNER, nullptr, xz,
            NTOK, 2 * D_INNER, D_MODEL);

        conv_silu_k<<<(NTOK * D_INNER) / 256, 256, 0, stream>>>(
            xz, convW + (size_t)i * D_CONV * D_INNER, convB + i * D_INNER, xc);

        wmma_gemm_f32<false, false><<<dim3(XD_W / 16, NTOK / 64), wblk, 0, stream>>>(
            xc, xpW + (size_t)i * D_INNER * XD_W, nullptr, xd, NTOK, XD_W, D_INNER);

        ssm_scan_k<<<dim3(D_INNER / 256, BATCH), 256, 0, stream>>>(
            xd, xz, dtW + (size_t)i * DT_RANK * D_INNER, dtB + i * D_INNER,
            A_log + (size_t)i * D_INNER * D_STATE, Dp + i * D_INNER, xc);

        // h += g @ outW
        wmma_gemm_f32<false, true><<<dim3(D_MODEL / 16, NTOK / 64), wblk, 0, stream>>>(
            xc, outW + (size_t)i * D_INNER * D_MODEL, nullptr, hbuf,
            NTOK, D_MODEL, D_INNER);
    }

    head_k<<<BATCH, 128, 0, stream>>>(hbuf, fln_w, fln_b, h1W, h1b, h2W, h2b, out);
}